// SinkhornOptimizer_v5_75436805587075
// MI455X (gfx1250) — compile-verified
//
#include <hip/hip_runtime.h>

// Log-stabilized Sinkhorn (POT sinkhorn_stabilized), batch of 128 independent
// 512x512 cost matrices, a=b=ones, reg=0.05, tau=1000, 50 iterations.
//
// One workgroup (1024 threads = 32 wave32) per batch element.
// Pass1 (colsum -> v): coalesced thread-per-column streaming.
// Pass2 (rowsum -> u): V_WMMA_F32_16X16X4_F32 with all-ones B as an in-pipe
//                      row-reduction tree (16 rows per wave).

#define SK_N     512
#define SK_REG   0.05f
#define SK_TAU   1000.0f
#define SK_ITERS 50

typedef __attribute__((ext_vector_type(2))) float v2f;
typedef __attribute__((ext_vector_type(8))) float v8f;

__global__ __launch_bounds__(1024)
void sinkhorn_stab_kernel(const float* __restrict__ X, float* __restrict__ Out)
{
    const int b = blockIdx.x;
    const float* __restrict__ M = X   + (size_t)b * SK_N * SK_N;
    float*       __restrict__ G = Out + (size_t)b * SK_N * SK_N;

    __shared__ float s_alpha[SK_N];
    __shared__ float s_beta [SK_N];
    __shared__ float s_u    [SK_N];
    __shared__ float s_v    [SK_N];
    __shared__ float s_part [2][SK_N];   // pass-1 partials; later log2(u)/log2(v)
    __shared__ int   s_imax;

    const int t    = threadIdx.x;
    const int lane = t & 31;
    const int wave = t >> 5;

    const float K2     = 1.0f / (SK_REG * 0.6931471805599453f); // 1/(reg*ln2): exp(x/reg)=exp2(x*K2)
    const float REGLN2 = SK_REG * 0.6931471805599453f;          // reg*ln2: reg*ln(x)=REGLN2*log2(x)
    const float invN   = 1.0f / (float)SK_N;

    if (t < SK_N) {
        s_alpha[t] = 0.0f;
        s_beta [t] = 0.0f;
        s_u    [t] = invN;
        s_v    [t] = invN;
    }
    if (t == 0) s_imax = 0;
    __syncthreads();

    for (int it = 0; it < SK_ITERS; ++it) {
        // ---------- Pass 1: colsum_j = sum_i K_ij * u_i ; v = 1/colsum ----------
        {
            const int   j  = t & (SK_N - 1);
            const int   h  = t >> 9;                 // row-half 0/1
            const float bj = s_beta[j];
            const int   i0 = h * (SK_N / 2);
            float acc = 0.0f;
            #pragma unroll 4
            for (int i = i0; i < i0 + SK_N / 2; ++i) {
                const float m = M[(size_t)i * SK_N + j];               // coalesced
                const float k = __builtin_amdgcn_exp2f((s_alpha[i] + bj - m) * K2);
                acc = fmaf(k, s_u[i], acc);
            }
            s_part[h][j] = acc;
        }
        __syncthreads();
        if (t < SK_N)
            s_v[t] = 1.0f / (s_part[0][t] + s_part[1][t]);
        __syncthreads();

        // ---------- Pass 2: rowsum_i = sum_j K_ij * v_j via WMMA ; u = 1/rowsum ----
        {
            const int   r0  = wave * 16;             // 32 waves * 16 rows = 512
            const int   row = r0 + (lane & 15);
            const int   jh  = (lane >> 4) * 2;       // lanes 0-15: K=0,1 ; lanes 16-31: K=2,3
            const float ar  = s_alpha[row];
            const float* __restrict__ Mrow = M + (size_t)row * SK_N;

            v8f acc = {};                            // C = 0
            v2f onesB; onesB.x = 1.0f; onesB.y = 1.0f;

            for (int j0 = 0; j0 < SK_N; j0 += 4) {
                const int j = j0 + jh;
                const v2f m = *(const v2f*)(Mrow + j);
                v2f a;
                a.x = __builtin_amdgcn_exp2f((ar + s_beta[j]     - m.x) * K2) * s_v[j];
                a.y = __builtin_amdgcn_exp2f((ar + s_beta[j + 1] - m.y) * K2) * s_v[j + 1];
                // D += A(16x4) x ones(4x16): every column of D = running row sums
                acc = __builtin_amdgcn_wmma_f32_16x16x4_f32(
                        false, a, false, onesB, (short)0, acc, false, false);
            }
            // D layout: VGPR d: lanes 0-15 -> (M=d, N=lane); lanes 16-31 -> (M=8+d)
            if (lane == 0) {
                #pragma unroll
                for (int d = 0; d < 8; ++d) s_u[r0 + d]     = 1.0f / acc[d];
            } else if (lane == 16) {
                #pragma unroll
                for (int d = 0; d < 8; ++d) s_u[r0 + 8 + d] = 1.0f / acc[d];
            }
        }
        __syncthreads();

        // ---------- absorption: (max|u| > tau) | (max|v| > tau) ----------
        {
            float mx = 0.0f;
            if (t < SK_N) mx = fmaxf(fabsf(s_u[t]), fabsf(s_v[t]));
            #pragma unroll
            for (int o = 16; o > 0; o >>= 1)
                mx = fmaxf(mx, __shfl_xor(mx, o, 32));
            if (lane == 0) atomicMax(&s_imax, __float_as_int(mx)); // positive floats: int order == float order
        }
        __syncthreads();
        const bool absorb = __int_as_float(s_imax) > SK_TAU;
        __syncthreads();                     // all reads of s_imax done
        if (t == 0) s_imax = 0;
        if (absorb && t < SK_N) {
            s_alpha[t] += REGLN2 * __builtin_amdgcn_logf(s_u[t]);  // reg*ln(u)
            s_beta [t] += REGLN2 * __builtin_amdgcn_logf(s_v[t]);  // reg*ln(v)
            s_u[t] = invN;
            s_v[t] = invN;
        }
        __syncthreads();
    }

    // ---------- Gamma = exp2((alpha_i+beta_j-M)*K2 + log2(u_i) + log2(v_j)) ----------
    if (t < SK_N) {
        s_part[0][t] = __builtin_amdgcn_logf(s_u[t]);   // log2(u)
        s_part[1][t] = __builtin_amdgcn_logf(s_v[t]);   // log2(v)
    }
    __syncthreads();
    {
        const int   j   = t & (SK_N - 1);
        const int   h   = t >> 9;
        const float bj  = s_beta[j];
        const float lvj = s_part[1][j];
        const int   i0  = h * (SK_N / 2);
        #pragma unroll 4
        for (int i = i0; i < i0 + SK_N / 2; ++i) {
            const float m = M[(size_t)i * SK_N + j];
            G[(size_t)i * SK_N + j] =
                __builtin_amdgcn_exp2f((s_alpha[i] + bj - m) * K2 + s_part[0][i] + lvj);
        }
    }
}

extern "C" void kernel_launch(void* const* d_in, const int* in_sizes, int n_in,
                              void* d_out, int out_size, void* d_ws, size_t ws_size,
                              hipStream_t stream)
{
    (void)n_in; (void)out_size; (void)d_ws; (void)ws_size;
    const float* X   = (const float*)d_in[0];
    float*       Out = (float*)d_out;
    const int batch = in_sizes[0] / (SK_N * SK_N);   // 128
    sinkhorn_stab_kernel<<<batch, 1024, 0, stream>>>(X, Out);
}